// DeformableConv2d_58557584113705
// MI455X (gfx1250) — compile-verified
//
#include <hip/hip_runtime.h>
#include <math.h>

typedef __attribute__((ext_vector_type(2))) float v2f;
typedef __attribute__((ext_vector_type(8))) float v8f;

#define KK      9             // 3x3 taps
#define CIN     128
#define OUTC    128
#define HDIM    128
#define WDIM    128
#define BATCH   4
#define TILE_N  32            // output pixels per block (along W)
#define CCHUNK  16            // input channels per K-chunk
#define KCHUNK  (CCHUNK*KK)   // 144
#define NCHUNKS (CIN/CCHUNK)  // 8
#define NOFF    18            // offset channels (9 taps x {dy,dx})
#define NOM     27            // offset + mask channels
#define PCOLS   (TILE_N+2)    // 34 patch columns
#define PSZ     (CCHUNK*3*PCOLS)   // 1632
#define NTAPPX  (KK*TILE_N)   // 288 (kk,n) combos per block

__global__ __launch_bounds__(256)
void dcn_fused_kernel(const float* __restrict__ x,
                      const float* __restrict__ w_off,
                      const float* __restrict__ b_off,
                      const float* __restrict__ w_mod,
                      const float* __restrict__ b_mod,
                      const float* __restrict__ weight,
                      const float* __restrict__ bias,
                      float* __restrict__ out)
{
    __shared__ float  lds_w[NOM * KCHUNK];       // 3888 f  offset/mask conv weights (per chunk)
    __shared__ float  lds_patch[PSZ];            // 1632 f  x patch (per chunk)
    __shared__ float  lds_om[NOM * TILE_N];      //  864 f  dy/dx/mask per pixel
    __shared__ int4   lds_sidx[NTAPPX];          // 4 clamped flat offsets per (kk,n)
    __shared__ float4 lds_swt[NTAPPX];           // 4 fused bilinear*mask*inb weights
    __shared__ float  lds_S[2][KCHUNK * TILE_N]; // double-buffered im2col chunk (144x32)

    const int t   = threadIdx.x;
    const int ow0 = blockIdx.x * TILE_N;
    const int oh  = blockIdx.y;
    const int b   = blockIdx.z;

    const float* xb = x + (size_t)b * CIN * HDIM * WDIM;

    // ---------------- Phase 1: 27-channel offset/modulation conv ----------------
    // thread -> (pixel px, base out-channel oc0); 4 accumulators: oc0, +8, +16, +24
    const int px  = t & 31;        // pixel in tile
    const int oc0 = t >> 5;        // 0..7  (wave-uniform -> broadcast weight reads)
    float a0 = 0.f, a1 = 0.f, a2 = 0.f, a3 = 0.f;

    for (int cc = 0; cc < NCHUNKS; ++cc) {
        for (int widx = t; widx < NOM * KCHUNK; widx += 256) {
            int och = widx / KCHUNK;
            int r   = widx - och * KCHUNK;        // c_local*9 + kk
            int cl  = r / KK, kk = r - cl * KK;
            int c   = cc * CCHUNK + cl;
            lds_w[widx] = (och < NOFF)
                ? w_off[((size_t)och * CIN + c) * KK + kk]
                : w_mod[((size_t)(och - NOFF) * CIN + c) * KK + kk];
        }
        for (int idx = t; idx < PSZ; idx += 256) {
            int cl  = idx / (3 * PCOLS);
            int rem = idx - cl * (3 * PCOLS);
            int row = rem / PCOLS, col = rem - row * PCOLS;
            int gy = oh - 1 + row, gx = ow0 - 1 + col;
            int c  = cc * CCHUNK + cl;
            float v = 0.f;
            if (gy >= 0 && gy < HDIM && gx >= 0 && gx < WDIM)
                v = xb[((size_t)c * HDIM + gy) * WDIM + gx];
            lds_patch[idx] = v;
        }
        __syncthreads();
        for (int cl = 0; cl < CCHUNK; ++cl) {
            const float* pp = &lds_patch[cl * 3 * PCOLS];
            const float* pw = &lds_w[cl * KK];
            #pragma unroll
            for (int kk = 0; kk < KK; ++kk) {
                int row = kk / 3, colk = kk - row * 3;
                float xv = pp[row * PCOLS + px + colk];
                a0 = fmaf(xv, pw[(oc0     ) * KCHUNK + kk], a0);
                a1 = fmaf(xv, pw[(oc0 +  8) * KCHUNK + kk], a1);
                a2 = fmaf(xv, pw[(oc0 + 16) * KCHUNK + kk], a2);
                if (oc0 + 24 < NOM)
                    a3 = fmaf(xv, pw[(oc0 + 24) * KCHUNK + kk], a3);
            }
        }
        __syncthreads();
    }
    // publish dy/dx (raw) and mask (sigmoid) per pixel
    {
        float accs[4] = {a0, a1, a2, a3};
        #pragma unroll
        for (int j = 0; j < 4; ++j) {
            int och = oc0 + j * 8;
            if (och < NOFF) {
                lds_om[och * TILE_N + px] = accs[j] + b_off[och];
            } else if (och < NOM) {
                float v = accs[j] + b_mod[och - NOFF];
                lds_om[och * TILE_N + px] = 1.0f / (1.0f + __expf(-v));
            }
        }
    }
    __syncthreads();

    // ---------------- Precompute bilinear tables: one entry per (kk, n) ----------------
    for (int p = t; p < NTAPPX; p += 256) {
        int kk = p >> 5;           // 0..8
        int n  = p & 31;
        int ky = kk / 3 - 1;
        int kx = kk - (kk / 3) * 3 - 1;
        float dy = lds_om[(kk * 2)     * TILE_N + n];
        float dx = lds_om[(kk * 2 + 1) * TILE_N + n];
        float m  = lds_om[(NOFF + kk)  * TILE_N + n];
        float py  = (float)(oh + ky) + dy;
        float pxx = (float)(ow0 + n + kx) + dx;
        float fy = floorf(py), fx = floorf(pxx);
        int y0 = (int)fy, x0 = (int)fx;
        float ly = py - fy, lx = pxx - fx;
        int y0c = min(max(y0, 0), HDIM - 1), y1c = min(max(y0 + 1, 0), HDIM - 1);
        int x0c = min(max(x0, 0), WDIM - 1), x1c = min(max(x0 + 1, 0), WDIM - 1);
        float iy0 = (y0 >= 0 && y0 < HDIM) ? 1.f : 0.f;
        float iy1 = (y0 + 1 >= 0 && y0 + 1 < HDIM) ? 1.f : 0.f;
        float ix0 = (x0 >= 0 && x0 < WDIM) ? 1.f : 0.f;
        float ix1 = (x0 + 1 >= 0 && x0 + 1 < WDIM) ? 1.f : 0.f;
        lds_sidx[p] = make_int4(y0c * WDIM + x0c, y0c * WDIM + x1c,
                                y1c * WDIM + x0c, y1c * WDIM + x1c);
        lds_swt[p]  = make_float4((1.f - ly) * (1.f - lx) * m * iy0 * ix0,
                                  (1.f - ly) * lx         * m * iy0 * ix1,
                                  ly * (1.f - lx)         * m * iy1 * ix0,
                                  ly * lx                 * m * iy1 * ix1);
    }
    __syncthreads();

    // ---------------- Phase 2: deformed gather + WMMA GEMM (double-buffered S) ----------------
    // 8 waves = 4 M-slabs (32 rows) x 2 N-tiles (16 px); two WMMAs share one B tile.
    const int wave  = t >> 5;
    const int lane  = t & 31;
    const int ln16  = lane & 15;
    const int hi    = lane >> 4;          // 0: K 0,1 / M 0..7   1: K 2,3 / M 8..15
    const int mslab = wave & 3;
    const int ntile = wave >> 2;
    const int obase = mslab << 5;         // 32 output rows per wave
    const int ncol  = (ntile << 4) + ln16;

    v8f acc0 = {}, acc1 = {};
    #pragma unroll
    for (int r = 0; r < 8; ++r) {
        acc0[r] = bias[obase +      r + hi * 8];
        acc1[r] = bias[obase + 16 + r + hi * 8];
    }

    for (int cc = 0; cc < NCHUNKS; ++cc) {
        const int buf = cc & 1;
        // build S[k_local (144), n (32)] = sum4(x[corner] * fused_weight)
        for (int i = 0; i < (KCHUNK * TILE_N) / 256; ++i) {   // 18 entries/thread
            int e  = t + i * 256;
            int kl = e >> 5;                 // 0..143
            int n  = e & 31;
            int cl = kl / KK;
            int kk = kl - cl * KK;
            int p  = (kk << 5) | n;
            int4   id = lds_sidx[p];
            float4 w4 = lds_swt[p];
            const float* xc = xb + (size_t)(cc * CCHUNK + cl) * HDIM * WDIM;
            float sv = fmaf(xc[id.x], w4.x,
                       fmaf(xc[id.y], w4.y,
                       fmaf(xc[id.z], w4.z, xc[id.w] * w4.w)));
            lds_S[buf][kl * TILE_N + n] = sv;
        }
        __syncthreads();   // S(buf) ready; prior reads of this buffer retired >=1 barrier ago

        // 36 K-steps; each: 2 A-loads (b64) + 1 B pair -> 2 x V_WMMA_F32_16X16X4_F32
        const float* wrow0 = weight + (size_t)(obase + ln16) * (CIN * KK)
                                    + cc * KCHUNK + hi * 2;
        const float* wrow1 = wrow0 + (size_t)16 * (CIN * KK);
        const float* sb = lds_S[buf];
        #pragma unroll 4
        for (int s = 0; s < KCHUNK / 4; ++s) {
            v2f aA = *(const v2f*)(wrow0 + s * 4);   // 8B aligned
            v2f aB = *(const v2f*)(wrow1 + s * 4);
            v2f bm;
            bm.x = sb[(s * 4 + hi * 2 + 0) * TILE_N + ncol];
            bm.y = sb[(s * 4 + hi * 2 + 1) * TILE_N + ncol];
            acc0 = __builtin_amdgcn_wmma_f32_16x16x4_f32(
                       false, aA, false, bm, (short)0, acc0, false, false);
            acc1 = __builtin_amdgcn_wmma_f32_16x16x4_f32(
                       false, aB, false, bm, (short)0, acc1, false, false);
        }
    }

    // store: D VGPR r -> rows base + r (lanes 0-15) / base + r + 8 (lanes 16-31)
    float* orow = out + (((size_t)b * OUTC) * HDIM + oh) * WDIM + ow0 + ncol;
    #pragma unroll
    for (int r = 0; r < 8; ++r) {
        int o0 = obase + r + hi * 8;
        orow[(size_t)o0 * HDIM * WDIM] = acc0[r];
        orow[(size_t)(o0 + 16) * HDIM * WDIM] = acc1[r];
    }
}

extern "C" void kernel_launch(void* const* d_in, const int* in_sizes, int n_in,
                              void* d_out, int out_size, void* d_ws, size_t ws_size,
                              hipStream_t stream) {
    const float* x      = (const float*)d_in[0];
    const float* w_off  = (const float*)d_in[1];
    const float* b_off  = (const float*)d_in[2];
    const float* w_mod  = (const float*)d_in[3];
    const float* b_mod  = (const float*)d_in[4];
    const float* weight = (const float*)d_in[5];
    const float* bias   = (const float*)d_in[6];
    float* out = (float*)d_out;

    dim3 grid(WDIM / TILE_N, HDIM, BATCH);   // 4 x 128 x 4 = 2048 blocks, 256 thr (8 waves)
    dcn_fused_kernel<<<grid, 256, 0, stream>>>(x, w_off, b_off, w_mod, b_mod,
                                               weight, bias, out);
}